// AnchorGCNLayer_34986803593487
// MI455X (gfx1250) — compile-verified
//
#include <hip/hip_runtime.h>

typedef __attribute__((ext_vector_type(2))) float v2f;
typedef __attribute__((ext_vector_type(8))) float v8f;

#define NN 50000
#define AA 500
#define FF 256
#define APAD 512
#define EPSF 1e-12f

__device__ __forceinline__ v8f wmma_f32(v2f a, v2f b, v8f c) {
  // D = A(16x4) * B(4x16) + C(16x16), fp32, wave32
  return __builtin_amdgcn_wmma_f32_16x16x4_f32(false, a, false, b, (short)0, c,
                                               false, false);
}

// -------- inv row sums: one wave per row ---------------------------------
__global__ __launch_bounds__(256) void k_rowsum_inv(const float* __restrict__ adj,
                                                    float* __restrict__ invrow) {
  const int lane = threadIdx.x & 31;
  const int wave = threadIdx.x >> 5;
  const int row = blockIdx.x * 8 + wave;   // grid = 6250 -> exactly 50000 rows
  const float* rp = adj + row * AA;        // 32-bit indexing (adj < 2GB)
  float s = 0.f;
  for (int c = lane; c < AA; c += 32) s += rp[c];
  for (int off = 16; off > 0; off >>= 1) s += __shfl_down(s, off, 32);
  if (lane == 0) invrow[row] = 1.0f / (s + EPSF);
}

// -------- column sums: per-block column partials + global atomics --------
__global__ __launch_bounds__(256) void k_colsum_part(const float* __restrict__ adj,
                                                     float* __restrict__ colsum) {
  const int t = threadIdx.x;
  const int r0 = blockIdx.x * 64;
  const int rend = (r0 + 64 < NN) ? (r0 + 64) : NN;
  const int c1 = t + 256;                  // second column per thread (t < 244)
  float a0 = 0.f, a1 = 0.f;
  for (int r = r0; r < rend; ++r) {
    const float* rp = adj + r * AA;
    a0 += rp[t];                           // coalesced across threads
    if (c1 < AA) a1 += rp[c1];
  }
  atomicAdd(&colsum[t], a0);
  if (c1 < AA) atomicAdd(&colsum[c1], a1);
}

__global__ void k_colsum_inv(const float* __restrict__ colsum,
                             float* __restrict__ invcol) {
  const int a = blockIdx.x * blockDim.x + threadIdx.x;
  if (a < APAD) invcol[a] = (a < AA) ? 1.0f / (colsum[a] + EPSF) : 0.0f;
}

// -------- msg0[a,f] += sum_n adj[n,a]*input[n,f]  (split-K, atomics) -----
// grid = 8 a-groups (64 rows) x 50 n-chunks (1000 rows) = 400 blocks x 512
__global__ __launch_bounds__(512) void k_atx(const float* __restrict__ adj,
                                             const float* __restrict__ inp,
                                             float* __restrict__ msg0) {
  // pair-interleaved A panel: element (n,a) at [(n>>1)*160 + a*2 + (n&1)]
  // pair-row pitch 160 == 32 (mod 64) -> the two lane halves hit disjoint
  // 32-bank sets on ds_load_b64, conflict-free.
  __shared__ float Alt[50 * 160];
  const int agroup = blockIdx.x & 7;
  const int chunk  = blockIdx.x >> 3;
  const int a0 = agroup * 64;
  const int nbase = chunk * 1000;
  const int wave = threadIdx.x >> 5;
  const int lane = threadIdx.x & 31;
  const int f0 = wave * 16;
  const int half = lane >> 4;
  const int mrow = lane & 15;
  v8f c0 = {}, c1 = {}, c2 = {}, c3 = {};
  const float* bbase = inp + (nbase + half * 2) * FF + f0 + mrow;
  const int afrag = half * 160 + mrow * 2;
  for (int kk = 0; kk < 1000; kk += 100) {
    __syncthreads();
    for (int e = threadIdx.x; e < 100 * 64; e += 512) {
      const int pn = e >> 6;
      const int pa = e & 63;
      const int a = a0 + pa;
      const float v = (a < AA) ? adj[(nbase + kk + pn) * AA + a] : 0.0f;
      Alt[(pn >> 1) * 160 + pa * 2 + (pn & 1)] = v;
    }
    __syncthreads();
    const float* bp = bbase + kk * FF;
#pragma unroll
    for (int s = 0; s < 25; ++s) {
      v2f b;
      b.x = bp[s * 4 * FF];                // immediate-offset global_load_b32
      b.y = bp[s * 4 * FF + FF];
      const float* ap = &Alt[afrag + s * 320];
      c0 = wmma_f32(*(const v2f*)(ap +  0), b, c0);   // ds_load_b64 each
      c1 = wmma_f32(*(const v2f*)(ap + 32), b, c1);
      c2 = wmma_f32(*(const v2f*)(ap + 64), b, c2);
      c3 = wmma_f32(*(const v2f*)(ap + 96), b, c3);
    }
  }
  const int col = f0 + mrow;
#pragma unroll
  for (int v = 0; v < 8; ++v) {
    const int m = v + half * 8;
    atomicAdd(&msg0[(a0 +  0 + m) * FF + col], c0[v]);
    atomicAdd(&msg0[(a0 + 16 + m) * FF + col], c1[v]);
    atomicAdd(&msg0[(a0 + 32 + m) * FF + col], c2[v]);
    atomicAdd(&msg0[(a0 + 48 + m) * FF + col], c3[v]);
  }
}

// -------- msg[a,f] = invcol[a] * sum_k msg0[a,k]*W[f,k] ------------------
// grid = 32 blocks (16 a-rows) x 512 threads (16 waves, one f-tile each)
__global__ __launch_bounds__(512) void k_msgw(const float* __restrict__ msg0,
                                              const float* __restrict__ Wm,
                                              const float* __restrict__ invcol,
                                              float* __restrict__ msg) {
  __shared__ float Als[16][260];           // pitch 260 -> banks (4m+k)%64
  const int a0 = blockIdx.x * 16;
  const int wave = threadIdx.x >> 5;
  const int lane = threadIdx.x & 31;
  const int f0 = wave * 16;
  const int half = lane >> 4;
  const int mrow = lane & 15;
  for (int e = threadIdx.x; e < 16 * 256; e += 512) {
    const int m = e >> 8, k = e & 255;
    Als[m][k] = msg0[(a0 + m) * FF + k] * invcol[a0 + m];  // scale folded in
  }
  __syncthreads();
  v8f c = {};
  const float* wb = Wm + (f0 + mrow) * FF + half * 2;  // B frag contiguous!
  const float* ab = &Als[mrow][half * 2];
#pragma unroll
  for (int s = 0; s < 64; ++s) {
    v2f b = *(const v2f*)(wb + s * 4);     // global_load_b64
    v2f a = *(const v2f*)(ab + s * 4);     // ds_load_b64
    c = wmma_f32(a, b, c);
  }
  const int col = f0 + mrow;
#pragma unroll
  for (int v = 0; v < 8; ++v)
    msg[(a0 + v + half * 8) * FF + col] = c[v];
}

// -------- out[n,f] = invrow[n] * sum_a adj[n,a]*msg[a,f] -----------------
// grid = 782 blocks x 512 threads (16 waves = 16 f-tiles, 4 m-tiles/wave)
__global__ __launch_bounds__(512) void k_out(const float* __restrict__ adj,
                                             const float* __restrict__ msg,
                                             const float* __restrict__ invrow,
                                             float* __restrict__ out) {
  extern __shared__ float Als[];           // [64][516]; frag pair contiguous in k
  const int LDP = 516;
  const int m0 = blockIdx.x * 64;
  const int wave = threadIdx.x >> 5;
  const int lane = threadIdx.x & 31;
  const int f0 = wave * 16;
  const int half = lane >> 4;
  const int mrow = lane & 15;
  const bool full = (m0 + 64 <= NN);
  if (full) {
    for (int e = threadIdx.x; e < 64 * 512; e += 512) {
      const int m = e >> 9, k = e & 511;
      Als[m * LDP + k] = (k < AA) ? adj[(m0 + m) * AA + k] : 0.0f;
    }
  } else {
    for (int e = threadIdx.x; e < 64 * 512; e += 512) {
      const int m = e >> 9, k = e & 511;
      const int gm = m0 + m;
      Als[m * LDP + k] = (k < AA && gm < NN) ? adj[gm * AA + k] : 0.0f;
    }
  }
  __syncthreads();
  v8f c0 = {}, c1 = {}, c2 = {}, c3 = {};
  const float* bb = msg + half * 2 * FF + f0 + mrow;
  const float* a0p = &Als[mrow * LDP + half * 2];
#pragma unroll 8
  for (int s = 0; s < 128; ++s) {          // K padded 500->512, guard-free
    v2f b;
    b.x = bb[s * 4 * FF];                  // msg rows 500..511 are zero
    b.y = bb[s * 4 * FF + FF];
    const float* ap = a0p + s * 4;
    c0 = wmma_f32(*(const v2f*)(ap),            b, c0);
    c1 = wmma_f32(*(const v2f*)(ap + 16 * LDP), b, c1);
    c2 = wmma_f32(*(const v2f*)(ap + 32 * LDP), b, c2);
    c3 = wmma_f32(*(const v2f*)(ap + 48 * LDP), b, c3);
  }
  const int col = f0 + mrow;
  if (full) {
#pragma unroll
    for (int v = 0; v < 8; ++v) {
      const int mi = v + half * 8;
      int r;
      r = m0 +  0 + mi; out[r * FF + col] = c0[v] * invrow[r];
      r = m0 + 16 + mi; out[r * FF + col] = c1[v] * invrow[r];
      r = m0 + 32 + mi; out[r * FF + col] = c2[v] * invrow[r];
      r = m0 + 48 + mi; out[r * FF + col] = c3[v] * invrow[r];
    }
  } else {
#pragma unroll
    for (int v = 0; v < 8; ++v) {
      const int mi = v + half * 8;
      int r;
      r = m0 +  0 + mi; if (r < NN) out[r * FF + col] = c0[v] * invrow[r];
      r = m0 + 16 + mi; if (r < NN) out[r * FF + col] = c1[v] * invrow[r];
      r = m0 + 32 + mi; if (r < NN) out[r * FF + col] = c2[v] * invrow[r];
      r = m0 + 48 + mi; if (r < NN) out[r * FF + col] = c3[v] * invrow[r];
    }
  }
}

extern "C" void kernel_launch(void* const* d_in, const int* in_sizes, int n_in,
                              void* d_out, int out_size, void* d_ws, size_t ws_size,
                              hipStream_t stream) {
  const float* inp = (const float*)d_in[0];  // [50000,256]
  const float* adj = (const float*)d_in[1];  // [50000,500]
  const float* Wm  = (const float*)d_in[2];  // [256,256]
  float* out = (float*)d_out;                // [50000,256]

  char* ws = (char*)d_ws;                    // ~1.25 MB used
  float* msg0   = (float*)(ws);                                  // 512*256
  float* msg    = (float*)(ws + (size_t)APAD * FF * 4);          // 512*256
  float* colsum = (float*)(ws + (size_t)2 * APAD * FF * 4);      // 512
  float* invcol = colsum + APAD;                                 // 512
  float* invrow = invcol + APAD;                                 // 50000

  hipMemsetAsync(msg0, 0, (size_t)APAD * FF * sizeof(float), stream);
  hipMemsetAsync(colsum, 0, APAD * sizeof(float), stream);

  k_rowsum_inv<<<NN / 8, 256, 0, stream>>>(adj, invrow);
  k_colsum_part<<<(NN + 63) / 64, 256, 0, stream>>>(adj, colsum);
  k_colsum_inv<<<2, 256, 0, stream>>>(colsum, invcol);
  k_atx<<<400, 512, 0, stream>>>(adj, inp, msg0);
  k_msgw<<<32, 512, 0, stream>>>(msg0, Wm, invcol, msg);
  k_out<<<(NN + 63) / 64, 512, (size_t)64 * 516 * sizeof(float), stream>>>(
      adj, msg, invrow, out);
}